// TransitionDown_65592740544739
// MI455X (gfx1250) — compile-verified
//
#include <hip/hip_runtime.h>
#include <stdint.h>

// Problem shape (fixed by the reference)
#define B_  8
#define N_  32768
#define C_  256
#define S_  8192
#define NT  1024          // FPS threads per block = 32 wave32s (max per WGP)
#define PPT (N_ / NT)     // 32 points per thread

// ---------------------------------------------------------------------------
// argmax combine with jnp.argmax tie-break (first == lowest index wins).
// Carries the winning point's coordinates so the next-iteration center never
// needs a dependent global load on the critical path.
// ---------------------------------------------------------------------------
__device__ __forceinline__ void combine(float& v, int& i, float& x, float& y, float& z,
                                        float ov, int oi, float ox, float oy, float oz) {
  if (ov > v || (ov == v && oi < i)) { v = ov; i = oi; x = ox; y = oy; z = oz; }
}

// wave32 butterfly argmax; leaves the reduction result in every lane.
// Lowers to ds_swizzle/ds_bpermute on gfx1250 (wave32).
__device__ __forceinline__ void wave_argmax(float& v, int& i, float& x, float& y, float& z) {
#pragma unroll
  for (int off = 16; off >= 1; off >>= 1) {
    const float ov = __shfl_xor(v, off);
    const int   oi = __shfl_xor(i, off);
    const float ox = __shfl_xor(x, off);
    const float oy = __shfl_xor(y, off);
    const float oz = __shfl_xor(z, off);
    combine(v, i, x, y, z, ov, oi, ox, oy, oz);
  }
}

// ---------------------------------------------------------------------------
// Kernel 1: farthest point sampling. One block per batch (the argmax is
// global over N, so per-batch work is inherently one-workgroup-sequential);
// min_dist lives in VGPRs (32 floats/thread); xyz (384KB/batch) streams from
// L2 (3MB total, fully L2-resident) every step.
//
// ONE barrier per iteration: per-wave winners go into a parity-double-
// buffered LDS array; after the barrier every wave redundantly reduces the
// 32 entries, so every thread knows the next center without a second barrier
// or an LDS broadcast. Double-buffering makes the single-barrier scheme
// race-free (a fast wave touches only the opposite-parity buffer and cannot
// reach reuse distance without crossing a barrier that orders all reads).
//
// Also: writes sampled_xyz directly, stashes the selected index in word 0 of
// the output feature row, and prefetches that row's 1KB into L2
// (global_prefetch_b8) so the final 64MB gather is L2-resident (192MB L2).
// ---------------------------------------------------------------------------
__global__ __launch_bounds__(NT, 1)
void fps_kernel(const float* __restrict__ xyz, const float* __restrict__ feat,
                float* __restrict__ out_xyz, float* __restrict__ out_feat) {
  const int b    = blockIdx.x;
  const int tid  = threadIdx.x;
  const int lane = tid & 31;
  const int wid  = tid >> 5;
  const float* xb = xyz + (size_t)b * N_ * 3;

  __shared__ float wv[2][32], wx[2][32], wy[2][32], wz[2][32];
  __shared__ int   wi[2][32];

  float md[PPT];
#pragma unroll
  for (int k = 0; k < PPT; ++k) md[k] = 1e10f;   // BIG from the reference

  // First center: index 0 (uniform scalar loads).
  int   ci = 0;
  float cx = xb[0], cy = xb[1], cz = xb[2];

  for (int s = 0; s < S_; ++s) {
    if (tid == 0) {
      const size_t row = (size_t)b * S_ + s;
      out_xyz[row * 3 + 0] = cx;
      out_xyz[row * 3 + 1] = cy;
      out_xyz[row * 3 + 2] = cz;
      ((int*)(out_feat + row * (size_t)C_))[0] = ci;   // stash index for gather
    }
    // Warm L2 with this row's feature data (8 lanes x 128B = 1KB row).
    if (tid < 8) {
      __builtin_prefetch(feat + ((size_t)b * N_ + ci) * C_ + tid * 32, 0, 3);
    }

    // Rank-1 distance update + thread-local argmax over this thread's points.
    float bv = -1.0f; int bi = 0;
#pragma unroll
    for (int k = 0; k < PPT; ++k) {
      const int p = tid + k * NT;
      const float* q = xb + 3 * (size_t)p;
      const float dx = q[0] - cx, dy = q[1] - cy, dz = q[2] - cz;
      const float d  = fmaf(dx, dx, fmaf(dy, dy, dz * dz));
      const float m  = fminf(md[k], d);
      md[k] = m;
      if (m > bv) { bv = m; bi = p; }   // k ascending == index ascending
    }
    // Reload the local winner's coords (L2-hot, overlaps the reduction).
    const float* qb = xb + 3 * (size_t)bi;
    float bx = qb[0], by = qb[1], bz = qb[2];

    wave_argmax(bv, bi, bx, by, bz);

    const int buf = s & 1;
    if (lane == 0) {
      wv[buf][wid] = bv; wi[buf][wid] = bi;
      wx[buf][wid] = bx; wy[buf][wid] = by; wz[buf][wid] = bz;
    }
    __syncthreads();   // the ONLY barrier per iteration

    // Every wave reduces the 32 per-wave winners (lane -> distinct entry,
    // bank-conflict-free); all threads end up with the global winner.
    bv = wv[buf][lane]; bi = wi[buf][lane];
    bx = wx[buf][lane]; by = wy[buf][lane]; bz = wz[buf][lane];
    wave_argmax(bv, bi, bx, by, bz);

    ci = bi; cx = bx; cy = by; cz = bz;
  }
}

// ---------------------------------------------------------------------------
// Kernel 2: feature gather via the CDNA5 async global<->LDS engine.
// One wave per 1KB row: 2x global_load_async_to_lds_b128 (32 lanes x 16B),
// s_wait_asynccnt, 2x global_store_async_from_lds_b128. GV addressing mode
// (per-lane 64-bit VGPR address, saddr=off) — no SGPR uniformity needed.
// The row's source index was stashed in word 0 of the destination row.
// Rows are L2-hot thanks to the prefetches issued during FPS.
// ---------------------------------------------------------------------------
__global__ __launch_bounds__(256)
void gather_feat_kernel(const float* __restrict__ feat, float* __restrict__ out_feat) {
  const int tid  = threadIdx.x;
  const int lane = tid & 31;
  const int w    = tid >> 5;
  const int row  = blockIdx.x * 8 + w;          // 65536 rows total
  const int b    = row >> 13;                   // row / S_

  __shared__ float lbuf[8 * C_];                // 1KB per wave
  float* lb = lbuf + (size_t)w * C_;

  float* dst = out_feat + (size_t)row * C_;
  const int idx = ((const int*)dst)[0];         // read BEFORE overwriting
  const float* src = feat + ((size_t)b * N_ + idx) * C_;

  // Low 32 bits of a generic shared pointer are the wave-relative LDS address.
  const uint32_t l0 = (uint32_t)(uintptr_t)lb + (uint32_t)lane * 16u;
  const uint32_t l1 = l0 + 512u;
  const uint64_t g0 = (uint64_t)(uintptr_t)src + (uint64_t)lane * 16u;
  const uint64_t g1 = g0 + 512u;

  asm volatile("global_load_async_to_lds_b128 %0, %1, off"
               :: "v"(l0), "v"(g0) : "memory");
  asm volatile("global_load_async_to_lds_b128 %0, %1, off"
               :: "v"(l1), "v"(g1) : "memory");
  __builtin_amdgcn_s_wait_asynccnt(0);

  const uint64_t d0 = (uint64_t)(uintptr_t)dst + (uint64_t)lane * 16u;
  const uint64_t d1 = d0 + 512u;
  asm volatile("global_store_async_from_lds_b128 %0, %1, off"
               :: "v"(d0), "v"(l0) : "memory");
  asm volatile("global_store_async_from_lds_b128 %0, %1, off"
               :: "v"(d1), "v"(l1) : "memory");
  __builtin_amdgcn_s_wait_asynccnt(0);
}

// ---------------------------------------------------------------------------
extern "C" void kernel_launch(void* const* d_in, const int* in_sizes, int n_in,
                              void* d_out, int out_size, void* d_ws, size_t ws_size,
                              hipStream_t stream) {
  (void)in_sizes; (void)n_in; (void)d_ws; (void)ws_size; (void)out_size;

  const float* xyz  = (const float*)d_in[0];   // (8, 32768, 3)   f32
  const float* feat = (const float*)d_in[1];   // (8, 32768, 256) f32

  float* out      = (float*)d_out;
  float* out_xyz  = out;                               // (8, 8192, 3)
  float* out_feat = out + (size_t)B_ * S_ * 3;         // (8, 8192, 256)

  fps_kernel<<<B_, NT, 0, stream>>>(xyz, feat, out_xyz, out_feat);
  gather_feat_kernel<<<(B_ * S_) / 8, 256, 0, stream>>>(feat, out_feat);
}